// SetApstractionLayer_60490319396908
// MI455X (gfx1250) — compile-verified
//
#include <hip/hip_runtime.h>

// ---------------------------------------------------------------------------
// PointNet++ Set Abstraction for MI455X (gfx1250, wave32, WMMA f32 16x16x4)
//
//   K1 fps_kernel        : 16 blocks (1/batch), points in LDS, 1024 argmax iters
//   K2 group_kernel      : 1 wave32 per centroid; ballot-based first-32-in-radius
//                          selection; writes grouped features G[8][S] (ch 6,7 = 0)
//   K3 conv1_stats       : WMMA conv1 (6->64), per-block partial sum/sumsq
//   K4 reduce_stats      : fixed-order reduction -> mean/rstd (BN1)
//   K5 conv2_stats       : recompute conv1+BN1+ReLU -> WMMA conv2 (64->128),
//                          per-block partial sum/sumsq
//   K6 reduce_stats      : -> mean/rstd (BN2)
//   K7 final_kernel      : recompute everything, BN2+ReLU, max over nsample=32
//
// Cross-lane reductions use DPP row_ror (VALU) instead of ds_bpermute.
// ---------------------------------------------------------------------------

#define BATCH 16
#define NPTS  4096
#define NCEN  1024
#define NSAMP 32
#define RAD2  0.25f
#define BNEPS 1e-5f

static constexpr size_t STOT = (size_t)BATCH * NCEN * NSAMP;   // 524288 samples
static constexpr int    NBLK = (int)(STOT / 128);              // 4096 conv blocks

typedef __attribute__((ext_vector_type(2))) float v2f;
typedef __attribute__((ext_vector_type(8))) float v8f;

__device__ __forceinline__ v8f wmma4(v2f a, v2f b, v8f c) {
  // D = A(16x4 f32) * B(4x16 f32) + C(16x16 f32)
  return __builtin_amdgcn_wmma_f32_16x16x4_f32(
      /*neg_a=*/false, a, /*neg_b=*/false, b,
      /*c_mod=*/(short)0, c, /*reuse_a=*/false, /*reuse_b=*/false);
}

// DPP row rotate within rows of 16 lanes (row_ror:N, ctrl = 0x120 + N).
template <int CTRL>
__device__ __forceinline__ float dpp_rot(float v) {
  return __int_as_float(__builtin_amdgcn_update_dpp(
      0, __float_as_int(v), CTRL, 0xf, 0xf, true));
}

// Sum/max over the 16 lanes of each half-wave row (all lanes get the result).
__device__ __forceinline__ float hsum16(float v) {
  v += dpp_rot<0x121>(v);  // row_ror:1
  v += dpp_rot<0x122>(v);  // row_ror:2
  v += dpp_rot<0x124>(v);  // row_ror:4
  v += dpp_rot<0x128>(v);  // row_ror:8
  return v;
}
__device__ __forceinline__ float hmax16(float v) {
  v = fmaxf(v, dpp_rot<0x121>(v));
  v = fmaxf(v, dpp_rot<0x122>(v));
  v = fmaxf(v, dpp_rot<0x124>(v));
  v = fmaxf(v, dpp_rot<0x128>(v));
  return v;
}

// ---------------------------------------------------------------------------
// K1: farthest point sampling, one block per batch. Points cached in LDS
// (48KB of the 320KB WGP LDS), per-thread min-distances in registers.
// ---------------------------------------------------------------------------
__global__ __launch_bounds__(256) void fps_kernel(const float* __restrict__ x,
                                                  float* __restrict__ centroids) {
  __shared__ float px[NPTS], py[NPTS], pz[NPTS];
  __shared__ float redv[8];
  __shared__ int   redi[8];
  __shared__ int   sbest;
  const int b = blockIdx.x, t = threadIdx.x;
  const float* xb = x + (size_t)b * NPTS * 3;
  for (int i = t; i < NPTS; i += 256) {
    px[i] = xb[i * 3 + 0];
    py[i] = xb[i * 3 + 1];
    pz[i] = xb[i * 3 + 2];
  }
  float dist[16];
#pragma unroll
  for (int i = 0; i < 16; ++i) dist[i] = 1e10f;
  __syncthreads();

  int farthest = 0;
  for (int it = 0; it < NCEN; ++it) {
    if (t == 0) {  // record BEFORE update (matches scan semantics)
      size_t o = ((size_t)b * NCEN + it) * 3;
      centroids[o + 0] = px[farthest];
      centroids[o + 1] = py[farthest];
      centroids[o + 2] = pz[farthest];
    }
    const float cx = px[farthest], cy = py[farthest], cz = pz[farthest];
    float bv = -1.f;
    int   bi = 0;
#pragma unroll
    for (int i = 0; i < 16; ++i) {
      const int p = t + (i << 8);
      const float dx = px[p] - cx, dy = py[p] - cy, dz = pz[p] - cz;
      const float d  = fmaf(dx, dx, fmaf(dy, dy, dz * dz));
      const float dm = fminf(dist[i], d);
      dist[i] = dm;
      if (dm > bv) { bv = dm; bi = p; }  // strict > keeps lowest index on tie
    }
#pragma unroll
    for (int off = 16; off > 0; off >>= 1) {
      const float ov = __shfl_down(bv, off);
      const int   oi = __shfl_down(bi, off);
      if (ov > bv || (ov == bv && oi < bi)) { bv = ov; bi = oi; }
    }
    if ((t & 31) == 0) { redv[t >> 5] = bv; redi[t >> 5] = bi; }
    __syncthreads();
    if (t == 0) {
      float best = redv[0];
      int besti = redi[0];
      for (int wv = 1; wv < 8; ++wv)
        if (redv[wv] > best || (redv[wv] == best && redi[wv] < besti)) {
          best = redv[wv]; besti = redi[wv];
        }
      sbest = besti;
    }
    __syncthreads();
    farthest = sbest;
  }
}

// ---------------------------------------------------------------------------
// K2: ball query + grouping. One wave32 per centroid (nsample==32==wave).
// First 32 in-radius indices in ascending order via ballot/prefix-popcount.
// Writes channel-major grouped features G[8][STOT]; channels 6,7 zeroed so
// conv1 K-padding needs no divergent guards.
// ---------------------------------------------------------------------------
__global__ __launch_bounds__(128) void group_kernel(const float* __restrict__ x,
                                                    const float* __restrict__ points,
                                                    const float* __restrict__ centroids,
                                                    float* __restrict__ G) {
  __shared__ int sel[4][NSAMP];
  const int wv = threadIdx.x >> 5, lane = threadIdx.x & 31;
  const int g = blockIdx.x * 4 + wv;      // global centroid id = b*1024 + s
  const int b = g >> 10;
  const float* xb = x + (size_t)b * NPTS * 3;
  const float cx = centroids[(size_t)g * 3 + 0];
  const float cy = centroids[(size_t)g * 3 + 1];
  const float cz = centroids[(size_t)g * 3 + 2];

  int count = 0;
  for (int base = 0; base < NPTS && count < NSAMP; base += 32) {
    const int p = base + lane;
    const float dx = xb[p * 3 + 0] - cx;
    const float dy = xb[p * 3 + 1] - cy;
    const float dz = xb[p * 3 + 2] - cz;
    const float d = dx * dx + dy * dy + dz * dz;
    const bool in = (d <= RAD2);
    const unsigned mask = (unsigned)__ballot(in);
    const int pos = count + __popc(mask & ((1u << lane) - 1u));
    if (in && pos < NSAMP) sel[wv][pos] = p;
    count += __popc(mask);
  }
  asm volatile("s_wait_dscnt 0" ::: "memory");  // cross-lane LDS visibility
  const int tcnt = count < NSAMP ? count : NSAMP;
  const int first = sel[wv][0];  // centroid itself always in radius -> tcnt>=1
  const int sidx = (lane < tcnt) ? sel[wv][lane] : first;

  const size_t Sb = (size_t)g * NSAMP + lane;
  const float gx = xb[sidx * 3 + 0] - cx;
  const float gy = xb[sidx * 3 + 1] - cy;
  const float gz = xb[sidx * 3 + 2] - cz;
  const float* pb = points + (size_t)b * NPTS * 3;
  G[0 * STOT + Sb] = gx;
  G[1 * STOT + Sb] = gy;
  G[2 * STOT + Sb] = gz;
  G[3 * STOT + Sb] = pb[sidx * 3 + 0];
  G[4 * STOT + Sb] = pb[sidx * 3 + 1];
  G[5 * STOT + Sb] = pb[sidx * 3 + 2];
  G[6 * STOT + Sb] = 0.f;   // K-dim padding for conv1 WMMA
  G[7 * STOT + Sb] = 0.f;
}

// ---------------------------------------------------------------------------
// WMMA helpers. Layout (ISA 7.12.2, f32 16x16x4):
//   A (16Mx4K):  lane L -> M = L&15, vgpr r -> K = r + 2*(L>>4)
//   B (4Kx16N):  lane L -> N = L&15, vgpr r -> K = r + 2*(L>>4)
//   C/D (16x16): vgpr r -> M = r + 8*(L>>4), N = L&15
// ---------------------------------------------------------------------------
__device__ __forceinline__ void conv1_tiles(const float* __restrict__ G,
                                            const float* W1p,  // LDS [64][8], cols 6,7 zero
                                            size_t S0, int half, int n15,
                                            v8f acc[4]) {
#pragma unroll
  for (int m = 0; m < 4; ++m) acc[m] = {};
#pragma unroll
  for (int ks = 0; ks < 2; ++ks) {
    v2f Bt;
#pragma unroll
    for (int r = 0; r < 2; ++r) {
      const int c = 4 * ks + r + 2 * half;
      Bt[r] = G[(size_t)c * STOT + S0 + n15];  // channels 6,7 are zero-filled
    }
#pragma unroll
    for (int m = 0; m < 4; ++m) {
      v2f At;
#pragma unroll
      for (int r = 0; r < 2; ++r)
        At[r] = W1p[(16 * m + n15) * 8 + 4 * ks + r + 2 * half];
      acc[m] = wmma4(At, Bt, acc[m]);
    }
  }
}

// conv1 + BN1 + ReLU -> per-wave a1 LDS tile, layout a1w[c*16 + n] (B-friendly)
// P1 layout: [0..63]=b1 [64..127]=gamma1 [128..191]=beta1 [192..255]=mean1 [256..319]=rstd1
__device__ __forceinline__ void compute_a1(const float* __restrict__ G,
                                           const float* W1p, const float* P1,
                                           float* a1w, size_t S0, int half, int n15) {
  v8f acc[4];
  conv1_tiles(G, W1p, S0, half, n15, acc);
#pragma unroll
  for (int m = 0; m < 4; ++m) {
#pragma unroll
    for (int r = 0; r < 8; ++r) {
      const int o = 16 * m + r + 8 * half;
      const float y = acc[m][r] + P1[o];
      const float a = P1[64 + o] * ((y - P1[192 + o]) * P1[256 + o]) + P1[128 + o];
      a1w[o * 16 + n15] = fmaxf(a, 0.f);
    }
  }
  asm volatile("s_wait_dscnt 0" ::: "memory");  // same-wave cross-lane LDS ordering
}

// Hoist all conv2 B-tiles (activations, 16 K-steps) into registers once.
__device__ __forceinline__ void load_btiles(const float* a1w, int half, int n15,
                                            v2f Bt[16]) {
#pragma unroll
  for (int k = 0; k < 16; ++k) {
#pragma unroll
    for (int r = 0; r < 2; ++r)
      Bt[k][r] = a1w[(4 * k + r + 2 * half) * 16 + n15];
  }
}

// One conv2 output M-tile: 16 K-steps of v_wmma_f32_16x16x4_f32
__device__ __forceinline__ v8f conv2_mtile(const float* W2s, const v2f Bt[16],
                                           int m, int half, int n15) {
  v8f acc = {};
#pragma unroll
  for (int k = 0; k < 16; ++k) {
    v2f At;
#pragma unroll
    for (int r = 0; r < 2; ++r)
      At[r] = W2s[(16 * m + n15) * 64 + 4 * k + r + 2 * half];
    acc = wmma4(At, Bt[k], acc);
  }
  return acc;
}

// ---------------------------------------------------------------------------
// K3: conv1 pass for BN1 statistics. 256 thr = 8 waves, 16 samples/wave.
// Partials: p1[blk*128 + ch] = sum, p1[blk*128 + 64 + ch] = sumsq.
// ---------------------------------------------------------------------------
__global__ __launch_bounds__(256) void conv1_stats_kernel(const float* __restrict__ G,
                                                          const float* __restrict__ w1,
                                                          const float* __restrict__ b1,
                                                          float* __restrict__ p1) {
  __shared__ float W1p[64 * 8];
  __shared__ float B1s[64];
  __shared__ float wsum[8][64];
  __shared__ float wsq[8][64];
  const int t = threadIdx.x, wv = t >> 5, lane = t & 31;
  const int half = lane >> 4, n15 = lane & 15;
  for (int i = t; i < 512; i += 256) {
    const int o = i >> 3, c = i & 7;
    W1p[i] = (c < 6) ? w1[o * 6 + c] : 0.f;
  }
  if (t < 64) B1s[t] = b1[t];
  __syncthreads();

  const size_t S0 = (size_t)blockIdx.x * 128 + wv * 16;
  v8f acc[4];
  conv1_tiles(G, W1p, S0, half, n15, acc);
#pragma unroll
  for (int m = 0; m < 4; ++m) {
#pragma unroll
    for (int r = 0; r < 8; ++r) {
      const int o = 16 * m + r + 8 * half;
      const float y = acc[m][r] + B1s[o];
      const float s = hsum16(y);
      const float q = hsum16(y * y);
      if (n15 == 0) { wsum[wv][o] = s; wsq[wv][o] = q; }
    }
  }
  __syncthreads();
  if (t < 64) {
    float s = 0.f, q = 0.f;
#pragma unroll
    for (int w = 0; w < 8; ++w) { s += wsum[w][t]; q += wsq[w][t]; }
    p1[(size_t)blockIdx.x * 128 + t] = s;
    p1[(size_t)blockIdx.x * 128 + 64 + t] = q;
  }
}

// ---------------------------------------------------------------------------
// K4/K6: fixed-order partial reduction -> st[ch]=mean, st[nch+ch]=rstd
// ---------------------------------------------------------------------------
__global__ void reduce_stats_kernel(const float* __restrict__ p, int nch, int nblocks,
                                    float* __restrict__ st) {
  const int t = threadIdx.x;
  if (t >= nch) return;
  float s = 0.f, q = 0.f;
  for (int bk = 0; bk < nblocks; ++bk) {
    s += p[(size_t)bk * 2 * nch + t];
    q += p[(size_t)bk * 2 * nch + nch + t];
  }
  const float inv = 1.0f / (float)STOT;
  const float mean = s * inv;
  const float var = q * inv - mean * mean;
  st[t] = mean;
  st[nch + t] = rsqrtf(var + BNEPS);
}

// ---------------------------------------------------------------------------
// K5: recompute conv1+BN1+ReLU, conv2 via WMMA, BN2 statistics partials.
// p2 layout: [blk*256 + ch]=sum, [blk*256 + 128 + ch]=sumsq
// ---------------------------------------------------------------------------
__global__ __launch_bounds__(256) void conv2_stats_kernel(
    const float* __restrict__ G, const float* __restrict__ w1, const float* __restrict__ b1,
    const float* __restrict__ gamma1, const float* __restrict__ beta1,
    const float* __restrict__ st1, const float* __restrict__ w2,
    const float* __restrict__ b2, float* __restrict__ p2) {
  __shared__ float W1p[64 * 8];
  __shared__ float W2s[128 * 64];
  __shared__ float P1[5 * 64];
  __shared__ float B2s[128];
  __shared__ float a1s[8][64 * 16];
  __shared__ float wsum[8][128];
  __shared__ float wsq[8][128];
  const int t = threadIdx.x, wv = t >> 5, lane = t & 31;
  const int half = lane >> 4, n15 = lane & 15;
  for (int i = t; i < 512; i += 256) {
    const int o = i >> 3, c = i & 7;
    W1p[i] = (c < 6) ? w1[o * 6 + c] : 0.f;
  }
  for (int i = t; i < 8192; i += 256) W2s[i] = w2[i];
  if (t < 64) {
    P1[t] = b1[t]; P1[64 + t] = gamma1[t]; P1[128 + t] = beta1[t];
    P1[192 + t] = st1[t]; P1[256 + t] = st1[64 + t];
  }
  if (t < 128) B2s[t] = b2[t];
  __syncthreads();

  const size_t S0 = (size_t)blockIdx.x * 128 + wv * 16;
  float* a1w = a1s[wv];
  compute_a1(G, W1p, P1, a1w, S0, half, n15);
  v2f Bt[16];
  load_btiles(a1w, half, n15, Bt);

#pragma unroll
  for (int m = 0; m < 8; ++m) {
    v8f acc = conv2_mtile(W2s, Bt, m, half, n15);
#pragma unroll
    for (int r = 0; r < 8; ++r) {
      const int o = 16 * m + r + 8 * half;
      const float y = acc[r] + B2s[o];
      const float s = hsum16(y);
      const float q = hsum16(y * y);
      if (n15 == 0) { wsum[wv][o] = s; wsq[wv][o] = q; }
    }
  }
  __syncthreads();
  if (t < 128) {
    float s = 0.f, q = 0.f;
#pragma unroll
    for (int w = 0; w < 8; ++w) { s += wsum[w][t]; q += wsq[w][t]; }
    p2[(size_t)blockIdx.x * 256 + t] = s;
    p2[(size_t)blockIdx.x * 256 + 128 + t] = q;
  }
}

// ---------------------------------------------------------------------------
// K7: full recompute, BN2+ReLU, max over nsample (32 = 2 wave-tiles), output.
// P2 layout: [0..127]=b2 [128..]=gamma2 [256..]=beta2 [384..]=mean2 [512..]=rstd2
// ---------------------------------------------------------------------------
__global__ __launch_bounds__(256) void final_kernel(
    const float* __restrict__ G, const float* __restrict__ w1, const float* __restrict__ b1,
    const float* __restrict__ gamma1, const float* __restrict__ beta1,
    const float* __restrict__ st1, const float* __restrict__ w2,
    const float* __restrict__ b2, const float* __restrict__ gamma2,
    const float* __restrict__ beta2, const float* __restrict__ st2,
    float* __restrict__ out2) {
  __shared__ float W1p[64 * 8];
  __shared__ float W2s[128 * 64];
  __shared__ float P1[5 * 64];
  __shared__ float P2[5 * 128];
  __shared__ float a1s[8][64 * 16];
  __shared__ float hg[8][128];
  const int t = threadIdx.x, wv = t >> 5, lane = t & 31;
  const int half = lane >> 4, n15 = lane & 15;
  for (int i = t; i < 512; i += 256) {
    const int o = i >> 3, c = i & 7;
    W1p[i] = (c < 6) ? w1[o * 6 + c] : 0.f;
  }
  for (int i = t; i < 8192; i += 256) W2s[i] = w2[i];
  if (t < 64) {
    P1[t] = b1[t]; P1[64 + t] = gamma1[t]; P1[128 + t] = beta1[t];
    P1[192 + t] = st1[t]; P1[256 + t] = st1[64 + t];
  }
  if (t < 128) {
    P2[t] = b2[t]; P2[128 + t] = gamma2[t]; P2[256 + t] = beta2[t];
    P2[384 + t] = st2[t]; P2[512 + t] = st2[128 + t];
  }
  __syncthreads();

  const size_t S0 = (size_t)blockIdx.x * 128 + wv * 16;
  float* a1w = a1s[wv];
  compute_a1(G, W1p, P1, a1w, S0, half, n15);
  v2f Bt[16];
  load_btiles(a1w, half, n15, Bt);

#pragma unroll
  for (int m = 0; m < 8; ++m) {
    v8f acc = conv2_mtile(W2s, Bt, m, half, n15);
#pragma unroll
    for (int r = 0; r < 8; ++r) {
      const int o = 16 * m + r + 8 * half;
      const float y = acc[r] + P2[o];
      float a = P2[128 + o] * ((y - P2[384 + o]) * P2[512 + o]) + P2[256 + o];
      a = fmaxf(a, 0.f);
      a = hmax16(a);  // max over this wave's 16 samples (half group)
      if (n15 == 0) hg[wv][o] = a;
    }
  }
  __syncthreads();
  // combine the two half-group waves, write out[b][s][o]
  for (int k = t; k < 512; k += 256) {
    const int gi = k >> 7, o = k & 127;
    const float v = fmaxf(hg[2 * gi][o], hg[2 * gi + 1][o]);
    out2[((size_t)blockIdx.x * 4 + gi) * 128 + o] = v;
  }
}

// ---------------------------------------------------------------------------
extern "C" void kernel_launch(void* const* d_in, const int* in_sizes, int n_in,
                              void* d_out, int out_size, void* d_ws, size_t ws_size,
                              hipStream_t stream) {
  const float* x      = (const float*)d_in[0];
  const float* points = (const float*)d_in[1];
  const float* w1     = (const float*)d_in[2];
  const float* b1     = (const float*)d_in[3];
  const float* gamma1 = (const float*)d_in[4];
  const float* beta1  = (const float*)d_in[5];
  const float* w2     = (const float*)d_in[6];
  const float* b2     = (const float*)d_in[7];
  const float* gamma2 = (const float*)d_in[8];
  const float* beta2  = (const float*)d_in[9];

  float* centroids = (float*)d_out;                             // [16,1024,3]
  float* out2      = (float*)d_out + (size_t)BATCH * NCEN * 3;  // [16,1024,128]

  char* w = (char*)d_ws;
  float* G   = (float*)w; w += (size_t)8 * STOT * sizeof(float);      // 16.8 MB
  float* p1  = (float*)w; w += (size_t)NBLK * 128 * sizeof(float);    // 2 MB
  float* p2  = (float*)w; w += (size_t)NBLK * 256 * sizeof(float);    // 4 MB
  float* st1 = (float*)w; w += 128 * sizeof(float);
  float* st2 = (float*)w; w += 256 * sizeof(float);

  fps_kernel<<<BATCH, 256, 0, stream>>>(x, centroids);
  group_kernel<<<BATCH * NCEN / 4, 128, 0, stream>>>(x, points, centroids, G);
  conv1_stats_kernel<<<NBLK, 256, 0, stream>>>(G, w1, b1, p1);
  reduce_stats_kernel<<<1, 64, 0, stream>>>(p1, 64, NBLK, st1);
  conv2_stats_kernel<<<NBLK, 256, 0, stream>>>(G, w1, b1, gamma1, beta1, st1, w2, b2, p2);
  reduce_stats_kernel<<<1, 128, 0, stream>>>(p2, 128, NBLK, st2);
  final_kernel<<<NBLK, 256, 0, stream>>>(G, w1, b1, gamma1, beta1, st1, w2, b2,
                                         gamma2, beta2, st2, out2);
}